// LossFunction_49615462203607
// MI455X (gfx1250) — compile-verified
//
#include <hip/hip_runtime.h>

typedef __attribute__((ext_vector_type(2))) float v2f;
typedef __attribute__((ext_vector_type(4))) float v4f;
typedef __attribute__((ext_vector_type(8))) float v8f;

#define HUBER_C 0.01f

// order-preserving float->uint key (monotonic for all floats)
__device__ __forceinline__ unsigned fkey(float f) {
    unsigned u = __float_as_uint(f);
    return (u & 0x80000000u) ? ~u : (u | 0x80000000u);
}
__device__ __forceinline__ float fval(unsigned k) {
    return __uint_as_float((k & 0x80000000u) ? (k & 0x7FFFFFFFu) : ~k);
}

__device__ __forceinline__ float huber(float x, float c) {
    return (x < c) ? (0.5f * x * x) : (c * x - 0.5f * c * c);
}

__device__ float block_reduce_sum(float v) {
    __shared__ float sm[256];
    sm[threadIdx.x] = v;
    __syncthreads();
    for (int s = blockDim.x >> 1; s > 0; s >>= 1) {
        if ((int)threadIdx.x < s) sm[threadIdx.x] += sm[threadIdx.x + s];
        __syncthreads();
    }
    return sm[0];
}

// ---------------------------------------------------------------------------
// init: colmin keys to +max, outputs to zero
__global__ void init_ws_kernel(unsigned* __restrict__ colmin, float* __restrict__ out, int total) {
    int i = blockIdx.x * blockDim.x + threadIdx.x;
    if (i < total) colmin[i] = 0xFFFFFFFFu;
    if (i < 2) out[i] = 0.0f;
}

// ---------------------------------------------------------------------------
// Pairwise squared distance via V_WMMA_F32_16X16X4_F32 + running row/col mins.
// rows (M) = tgt points, cols (N) = src_transformed points.
// A row m = [tx,ty,tz,||t||^2]; B col n = [-2sx,-2sy,-2sz,1]
//   => C[m][n] = ||t_m||^2 - 2 t_m . s_n ; then add ||s_n||^2 (lane-constant).
// B-side fragments + norms are staged in LDS once per block: the inner loop is
// pure LDS + WMMA (no global traffic, no recomputed norms).
__global__ __launch_bounds__(256) void dist_min_kernel(
    const float* __restrict__ src,   // (B,3,N) preds
    const float* __restrict__ tgt,   // (B,3,N) gts
    float* __restrict__ rowmin,      // (B,N) plain floats (exclusively owned)
    unsigned* __restrict__ colmin,   // (B,N) uint keys, global atomic min
    int N)
{
    __shared__ unsigned cmin[4096];   // 16 KB
    __shared__ v4f      colq[4096];   // 64 KB: {-2sx, -2sy, -2sz, ||s||^2}

    const int b    = blockIdx.y;
    const int tid  = threadIdx.x;
    const int wave = tid >> 5;
    const int lane = tid & 31;
    const int lmod = lane & 15;
    const int half = lane >> 4;

    const float* tb = tgt + (size_t)b * 3 * N;
    const float* sb = src + (size_t)b * 3 * N;

    // Stage per-column B-side data into LDS (once per block)
    for (int i = tid; i < N; i += blockDim.x) {
        cmin[i] = 0xFFFFFFFFu;
        const float s0 = sb[0 * N + i];
        const float s1 = sb[1 * N + i];
        const float s2 = sb[2 * N + i];
        v4f q;
        q.x = -2.0f * s0;
        q.y = -2.0f * s1;
        q.z = -2.0f * s2;
        q.w = s0 * s0 + s1 * s1 + s2 * s2;
        colq[i] = q;
    }
    __syncthreads();

    // A fragment: 16x4 f32. lane L: M = L%16; lanes<16 hold K={0,1}, lanes>=16 K={2,3}
    const int row = blockIdx.x * 128 + wave * 16 + lmod;
    const float t0 = tb[0 * N + row];
    const float t1 = tb[1 * N + row];
    const float t2 = tb[2 * N + row];
    const float xx = t0 * t0 + t1 * t1 + t2 * t2;
    v2f afrag;
    afrag.x = half ? t2 : t0;
    afrag.y = half ? xx : t1;

    v8f rmin;
#pragma unroll
    for (int r = 0; r < 8; ++r) rmin[r] = 3.402823466e38f;

    for (int ct = 0; ct < N; ct += 16) {
        const int col = ct + lmod;
        const v4f q = colq[col];              // ds_load_b128

        // B fragment: 4x16 f32. lane L: N = L%16; lanes<16 hold K={0,1}, lanes>=16 K={2,3}
        v2f bfrag;
        bfrag.x = half ? q.z : q.x;
        bfrag.y = half ? 1.0f : q.y;
        const float yy = q.w;

        v8f c = {};
        c = __builtin_amdgcn_wmma_f32_16x16x4_f32(
            /*neg_a=*/false, afrag, /*neg_b=*/false, bfrag,
            /*c_mod=*/(short)0, c, /*reuse_a=*/false, /*reuse_b=*/false);

        float cpart = 3.402823466e38f;
#pragma unroll
        for (int r = 0; r < 8; ++r) {
            const float d = c[r] + yy;     // full squared distance
            rmin[r] = fminf(rmin[r], d);   // running row min (over columns)
            cpart   = fminf(cpart, d);     // partial col min (over 8 rows)
        }
        atomicMin(&cmin[col], fkey(cpart)); // ds_min_u32
    }

    // Row mins: reduce across the 16-lane half (xor masks stay inside the half)
#pragma unroll
    for (int r = 0; r < 8; ++r) {
        float v = rmin[r];
        v = fminf(v, __shfl_xor(v, 1, 32));
        v = fminf(v, __shfl_xor(v, 2, 32));
        v = fminf(v, __shfl_xor(v, 4, 32));
        v = fminf(v, __shfl_xor(v, 8, 32));
        rmin[r] = v;
    }
    if (lmod == 0) {
        // C layout: lanes 0-15 VGPR r -> M=r ; lanes 16-31 VGPR r -> M=8+r
        const int rbase = blockIdx.x * 128 + wave * 16 + half * 8;
#pragma unroll
        for (int r = 0; r < 8; ++r)
            rowmin[(size_t)b * N + rbase + r] = rmin[r];
    }

    __syncthreads();
    for (int i = tid; i < N; i += blockDim.x)
        atomicMin(&colmin[(size_t)b * N + i], cmin[i]);
}

// ---------------------------------------------------------------------------
// gal = sum huber(rowmin) + sum huber(colmin)
__global__ void finalize_gal_kernel(const float* __restrict__ rowmin,
                                    const unsigned* __restrict__ colmin,
                                    float* __restrict__ out, int total)
{
    float acc = 0.0f;
    for (int i = blockIdx.x * blockDim.x + threadIdx.x; i < total;
         i += gridDim.x * blockDim.x) {
        acc += huber(rowmin[i], HUBER_C);
        acc += huber(fval(colmin[i]), HUBER_C);
    }
    float s = block_reduce_sum(acc);
    if (threadIdx.x == 0) atomicAdd(&out[1], s);
}

// ---------------------------------------------------------------------------
// keypoints_loss = sum((R @ src_kp + t - tgt_kp)^2)
__global__ void kp_loss_kernel(const float* __restrict__ skp,  // (B,3,KP)
                               const float* __restrict__ tkp,  // (B,3,KP)
                               const float* __restrict__ R,    // (B,3,3)
                               const float* __restrict__ t,    // (B,3)
                               float* __restrict__ out, int B, int KP)
{
    const int idx = blockIdx.x * blockDim.x + threadIdx.x;
    float acc = 0.0f;
    if (idx < B * KP) {
        const int b = idx / KP;
        const int p = idx - b * KP;
        const float* Rb = R + b * 9;
        const float* tb = t + b * 3;
        const float s0 = skp[(b * 3 + 0) * KP + p];
        const float s1 = skp[(b * 3 + 1) * KP + p];
        const float s2 = skp[(b * 3 + 2) * KP + p];
#pragma unroll
        for (int d = 0; d < 3; ++d) {
            const float v = Rb[d * 3 + 0] * s0 + Rb[d * 3 + 1] * s1 +
                            Rb[d * 3 + 2] * s2 + tb[d] -
                            tkp[(b * 3 + d) * KP + p];
            acc += v * v;
        }
    }
    float s = block_reduce_sum(acc);
    if (threadIdx.x == 0) atomicAdd(&out[0], s);
}

// ---------------------------------------------------------------------------
// knn term: sum((a-b)^2) / k
__global__ void knn_loss_kernel(const float* __restrict__ a,
                                const float* __restrict__ bptr,
                                float* __restrict__ out, int total, float invk)
{
    float acc = 0.0f;
    for (int i = blockIdx.x * blockDim.x + threadIdx.x; i < total;
         i += gridDim.x * blockDim.x) {
        const float d = a[i] - bptr[i];
        acc += d * d;
    }
    float s = block_reduce_sum(acc) * invk;
    if (threadIdx.x == 0) atomicAdd(&out[0], s);
}

// ---------------------------------------------------------------------------
extern "C" void kernel_launch(void* const* d_in, const int* in_sizes, int n_in,
                              void* d_out, int out_size, void* d_ws, size_t ws_size,
                              hipStream_t stream)
{
    const float* src_kp  = (const float*)d_in[0];
    const float* tgt_kp  = (const float*)d_in[1];
    const float* rot     = (const float*)d_in[2];
    const float* trans   = (const float*)d_in[3];
    const float* knn_a   = (const float*)d_in[4];
    const float* knn_b   = (const float*)d_in[5];
    const float* src_tr  = (const float*)d_in[7];
    const float* tgt     = (const float*)d_in[8];
    float* out = (float*)d_out;

    const int B  = in_sizes[2] / 9;                 // rotation (B,3,3)
    const int KP = in_sizes[0] / (3 * B);           // keypoints (B,3,KP)
    const int N  = in_sizes[7] / (3 * B);           // clouds (B,3,N)
    const int knn_total = in_sizes[4];              // B*3*KP*K
    const int K  = knn_total / (3 * B * KP);

    // workspace: [B*N uint colmin keys][B*N float rowmin]
    unsigned* colmin = (unsigned*)d_ws;
    float*    rowmin = (float*)((unsigned*)d_ws + (size_t)B * N);
    const int totalBN = B * N;

    init_ws_kernel<<<(totalBN + 255) / 256, 256, 0, stream>>>(colmin, out, totalBN);

    dim3 grid(N / 128, B);
    dist_min_kernel<<<grid, 256, 0, stream>>>(src_tr, tgt, rowmin, colmin, N);

    finalize_gal_kernel<<<128, 256, 0, stream>>>(rowmin, colmin, out, totalBN);

    kp_loss_kernel<<<(B * KP + 255) / 256, 256, 0, stream>>>(
        src_kp, tgt_kp, rot, trans, out, B, KP);

    int knn_blocks = (knn_total + 255) / 256;
    if (knn_blocks > 512) knn_blocks = 512;
    knn_loss_kernel<<<knn_blocks, 256, 0, stream>>>(
        knn_a, knn_b, out, knn_total, 1.0f / (float)K);
}